// BahdanauAttention_56014963474605
// MI455X (gfx1250) — compile-verified
//
#include <hip/hip_runtime.h>
#include <math.h>

typedef float v2f __attribute__((ext_vector_type(2)));
typedef float v8f __attribute__((ext_vector_type(8)));

#define B_ 8
#define T_ 128
#define S_ 256
#define H_ 512

__device__ __forceinline__ float fast_tanh(float x) {
#if __has_builtin(__builtin_amdgcn_tanhf)
  return __builtin_amdgcn_tanhf(x);
#else
  // tanh(x) = (e^{2x}-1)/(e^{2x}+1), via v_exp_f32 (exp2)
  float xc = fminf(fmaxf(x, -15.0f), 15.0f);
  float e = __builtin_amdgcn_exp2f(xc * 2.885390081777927f); // 2*log2(e)
  return (e - 1.0f) / (e + 1.0f);
#endif
}

__device__ __forceinline__ v8f wmma4(v2f a, v2f b, v8f c) {
  return __builtin_amdgcn_wmma_f32_16x16x4_f32(false, a, false, b, (short)0, c,
                                               false, false);
}

// ---------------------------------------------------------------------------
// C[M,N] = A[M,K] @ W[N,K]^T   (torch Linear convention)
// block = 128 threads (4 waves); wave tile = 16(M) x 64(N); grid=(M/16, N/256)
// Explicit double-buffer (two disjoint register sets, unroll-by-2) so each
// WMMA group's operands were loaded a full iteration earlier.
// K must be a multiple of 8.
// ---------------------------------------------------------------------------
__global__ __launch_bounds__(128) void linear_wmma(
    const float* __restrict__ A, const float* __restrict__ W,
    float* __restrict__ C, int M, int N, int K) {
  const int lane = threadIdx.x & 31;
  const int wave = threadIdx.x >> 5;
  const int lo = lane & 15;
  const int hi = lane >> 4;
  const int m0 = blockIdx.x * 16;
  const int n0 = blockIdx.y * 256 + wave * 64;

  v8f acc0 = {}, acc1 = {}, acc2 = {}, acc3 = {};
  const float* arow = A + (size_t)(m0 + lo) * K + 2 * hi;
  const float* w0 = W + (size_t)(n0 +  0 + lo) * K + 2 * hi;
  const float* w1 = W + (size_t)(n0 + 16 + lo) * K + 2 * hi;
  const float* w2 = W + (size_t)(n0 + 32 + lo) * K + 2 * hi;
  const float* w3 = W + (size_t)(n0 + 48 + lo) * K + 2 * hi;

#define LIN_LOAD(aV, p0, p1, p2, p3, off)    \
  aV = *(const v2f*)(arow + (off));          \
  p0 = *(const v2f*)(w0 + (off));            \
  p1 = *(const v2f*)(w1 + (off));            \
  p2 = *(const v2f*)(w2 + (off));            \
  p3 = *(const v2f*)(w3 + (off));
#define LIN_WMMA(aV, p0, p1, p2, p3)         \
  acc0 = wmma4(aV, p0, acc0);                \
  acc1 = wmma4(aV, p1, acc1);                \
  acc2 = wmma4(aV, p2, acc2);                \
  acc3 = wmma4(aV, p3, acc3);

  v2f a0, b00, b01, b02, b03;   // set 0
  v2f a1, b10, b11, b12, b13;   // set 1
  LIN_LOAD(a0, b00, b01, b02, b03, 0)
  LIN_LOAD(a1, b10, b11, b12, b13, 4)
  for (int k = 8; k < K; k += 8) {
    LIN_WMMA(a0, b00, b01, b02, b03)
    LIN_LOAD(a0, b00, b01, b02, b03, k)
    LIN_WMMA(a1, b10, b11, b12, b13)
    LIN_LOAD(a1, b10, b11, b12, b13, k + 4)
  }
  LIN_WMMA(a0, b00, b01, b02, b03)
  LIN_WMMA(a1, b10, b11, b12, b13)
#undef LIN_LOAD
#undef LIN_WMMA

#pragma unroll
  for (int j = 0; j < 8; ++j) {
    const size_t r = (size_t)(m0 + j + 8 * hi) * N + n0 + lo;
    C[r +  0] = acc0[j];
    C[r + 16] = acc1[j];
    C[r + 32] = acc2[j];
    C[r + 48] = acc3[j];
  }
}

// ---------------------------------------------------------------------------
// scores + masked softmax:
// attn[b,t,s] = softmax_s( mask( sum_h v[h]*tanh(qs[b,t,h]+hs[b,s,h]) ) )
// 1 block = 256 threads handles (b, 8 consecutive t); thread <-> s (S_=256)
// ---------------------------------------------------------------------------
#define TT 8
#define HC 32
#define HCP 33   // pad stride: 33 coprime with 64 banks -> conflict-free

__global__ __launch_bounds__(256) void scores_softmax(
    const float* __restrict__ qs,    // (B*T, H)
    const float* __restrict__ hs,    // (B*S, H)
    const float* __restrict__ v,     // (H)
    const int*   __restrict__ lens,  // (B)
    float* __restrict__ attn)        // (B*T, S)
{
  __shared__ float sh[S_ * HCP];   // 33 KB
  __shared__ float sq[TT * HC];
  __shared__ float sv[HC];
  __shared__ float red[256];

  const int tid = threadIdx.x;
  const int b  = blockIdx.x / (T_ / TT);
  const int t0 = (blockIdx.x % (T_ / TT)) * TT;

  float acc[TT];
#pragma unroll
  for (int tt = 0; tt < TT; ++tt) acc[tt] = 0.0f;

  for (int hc = 0; hc < H_; hc += HC) {
    // stage hs[b, 0:S, hc:hc+HC] into LDS (coalesced float4 reads)
    for (int idx = tid; idx < S_ * (HC / 4); idx += 256) {
      const int s  = idx / (HC / 4);
      const int h4 = (idx % (HC / 4)) * 4;
      const float4 d = *(const float4*)(hs + (size_t)(b * S_ + s) * H_ + hc + h4);
      float* dst = sh + s * HCP + h4;
      dst[0] = d.x; dst[1] = d.y; dst[2] = d.z; dst[3] = d.w;
    }
    if (tid < TT * HC) {
      const int tt = tid / HC, h = tid % HC;
      sq[tid] = qs[(size_t)(b * T_ + t0 + tt) * H_ + hc + h];
    }
    if (tid < HC) sv[tid] = v[hc + tid];
    __syncthreads();

    const float* hrow = sh + tid * HCP;   // this thread's s-row
#pragma unroll 4
    for (int h = 0; h < HC; ++h) {
      const float hsv = hrow[h];
      const float vh  = sv[h];
#pragma unroll
      for (int tt = 0; tt < TT; ++tt)
        acc[tt] += vh * fast_tanh(hsv + sq[tt * HC + h]);
    }
    __syncthreads();
  }

  const int len = lens[b];
#pragma unroll
  for (int tt = 0; tt < TT; ++tt) {
    const bool valid = (tid < len);
    const float score = valid ? acc[tt] : -__builtin_inff();
    red[tid] = score;
    __syncthreads();
    for (int off = 128; off > 0; off >>= 1) {
      if (tid < off) red[tid] = fmaxf(red[tid], red[tid + off]);
      __syncthreads();
    }
    const float mx = red[0];
    __syncthreads();
    const float e = valid
        ? __builtin_amdgcn_exp2f((score - mx) * 1.4426950408889634f) : 0.0f;
    red[tid] = e;
    __syncthreads();
    for (int off = 128; off > 0; off >>= 1) {
      if (tid < off) red[tid] += red[tid + off];
      __syncthreads();
    }
    const float sum = red[0];
    __syncthreads();
    attn[(size_t)(b * T_ + t0 + tt) * S_ + tid] = e / sum;
  }
}

// ---------------------------------------------------------------------------
// context: per batch, C[T,H] = attn[T,S] @ enc[S,H] (row-major B)
// grid = (T/16, H/256, B), 128 threads; explicit double-buffer (unroll-by-2).
// ---------------------------------------------------------------------------
__global__ __launch_bounds__(128) void ctx_wmma(
    const float* __restrict__ attn, const float* __restrict__ enc,
    float* __restrict__ C) {
  const int lane = threadIdx.x & 31;
  const int wave = threadIdx.x >> 5;
  const int lo = lane & 15;
  const int hi = lane >> 4;
  const int bb = blockIdx.z;
  const float* A  = attn + (size_t)bb * T_ * S_;
  const float* Bm = enc  + (size_t)bb * S_ * H_;
  float* Cb = C + (size_t)bb * T_ * H_;
  const int m0 = blockIdx.x * 16;
  const int n0 = blockIdx.y * 256 + wave * 64;

  v8f acc0 = {}, acc1 = {}, acc2 = {}, acc3 = {};
  const float* arow = A + (size_t)(m0 + lo) * S_ + 2 * hi;
  // per-fragment B column bases (lane-fixed): element (k) at brow[i] + k*H_
  const float* br0 = Bm + (size_t)(2 * hi) * H_ + n0 +  0 + lo;
  const float* br1 = Bm + (size_t)(2 * hi) * H_ + n0 + 16 + lo;
  const float* br2 = Bm + (size_t)(2 * hi) * H_ + n0 + 32 + lo;
  const float* br3 = Bm + (size_t)(2 * hi) * H_ + n0 + 48 + lo;

#define CTX_LOAD(aV, p0, p1, p2, p3, off)                                \
  aV = *(const v2f*)(arow + (off));                                      \
  p0.x = br0[(size_t)(off) * H_];  p0.y = br0[(size_t)((off) + 1) * H_]; \
  p1.x = br1[(size_t)(off) * H_];  p1.y = br1[(size_t)((off) + 1) * H_]; \
  p2.x = br2[(size_t)(off) * H_];  p2.y = br2[(size_t)((off) + 1) * H_]; \
  p3.x = br3[(size_t)(off) * H_];  p3.y = br3[(size_t)((off) + 1) * H_];
#define CTX_WMMA(aV, p0, p1, p2, p3)   \
  acc0 = wmma4(aV, p0, acc0);          \
  acc1 = wmma4(aV, p1, acc1);          \
  acc2 = wmma4(aV, p2, acc2);          \
  acc3 = wmma4(aV, p3, acc3);

  v2f a0, b00, b01, b02, b03;
  v2f a1, b10, b11, b12, b13;
  CTX_LOAD(a0, b00, b01, b02, b03, 0)
  CTX_LOAD(a1, b10, b11, b12, b13, 4)
  for (int k = 8; k < S_; k += 8) {
    CTX_WMMA(a0, b00, b01, b02, b03)
    CTX_LOAD(a0, b00, b01, b02, b03, k)
    CTX_WMMA(a1, b10, b11, b12, b13)
    CTX_LOAD(a1, b10, b11, b12, b13, k + 4)
  }
  CTX_WMMA(a0, b00, b01, b02, b03)
  CTX_WMMA(a1, b10, b11, b12, b13)
#undef CTX_LOAD
#undef CTX_WMMA

#pragma unroll
  for (int j = 0; j < 8; ++j) {
    const size_t r = (size_t)(m0 + j + 8 * hi) * H_ + n0 + lo;
    Cb[r +  0] = acc0[j];
    Cb[r + 16] = acc1[j];
    Cb[r + 32] = acc2[j];
    Cb[r + 48] = acc3[j];
  }
}

// ---------------------------------------------------------------------------
// out[M,H] = tanh( bias + ctx[M,H] @ Wout[:, 0:H]^T + q[M,H] @ Wout[:, H:2H]^T )
// grid = (M/16, H/256), 128 threads; double-buffered per K-pass.
// ---------------------------------------------------------------------------
__global__ __launch_bounds__(128) void out_wmma(
    const float* __restrict__ ctx, const float* __restrict__ q,
    const float* __restrict__ Wout, const float* __restrict__ bias,
    float* __restrict__ out) {
  const int lane = threadIdx.x & 31;
  const int wave = threadIdx.x >> 5;
  const int lo = lane & 15;
  const int hi = lane >> 4;
  const int m0 = blockIdx.x * 16;
  const int n0 = blockIdx.y * 256 + wave * 64;
  const int ld = 2 * H_;

  v8f acc0 = {}, acc1 = {}, acc2 = {}, acc3 = {};
  const float* a1p = ctx + (size_t)(m0 + lo) * H_ + 2 * hi;
  const float* a2p = q   + (size_t)(m0 + lo) * H_ + 2 * hi;
  const float* w0 = Wout + (size_t)(n0 +  0 + lo) * ld + 2 * hi;
  const float* w1 = Wout + (size_t)(n0 + 16 + lo) * ld + 2 * hi;
  const float* w2 = Wout + (size_t)(n0 + 32 + lo) * ld + 2 * hi;
  const float* w3 = Wout + (size_t)(n0 + 48 + lo) * ld + 2 * hi;

#define OUT_LOAD(ap, aV, p0, p1, p2, p3, aoff, woff) \
  aV = *(const v2f*)(ap + (aoff));                   \
  p0 = *(const v2f*)(w0 + (woff));                   \
  p1 = *(const v2f*)(w1 + (woff));                   \
  p2 = *(const v2f*)(w2 + (woff));                   \
  p3 = *(const v2f*)(w3 + (woff));
#define OUT_WMMA(aV, p0, p1, p2, p3)   \
  acc0 = wmma4(aV, p0, acc0);          \
  acc1 = wmma4(aV, p1, acc1);          \
  acc2 = wmma4(aV, p2, acc2);          \
  acc3 = wmma4(aV, p3, acc3);

  v2f a0, b00, b01, b02, b03;
  v2f a1, b10, b11, b12, b13;

  // ---- pass 1: context half of W_out (cols 0..H-1) ----
  OUT_LOAD(a1p, a0, b00, b01, b02, b03, 0, 0)
  OUT_LOAD(a1p, a1, b10, b11, b12, b13, 4, 4)
  for (int k = 8; k < H_; k += 8) {
    OUT_WMMA(a0, b00, b01, b02, b03)
    OUT_LOAD(a1p, a0, b00, b01, b02, b03, k, k)
    OUT_WMMA(a1, b10, b11, b12, b13)
    OUT_LOAD(a1p, a1, b10, b11, b12, b13, k + 4, k + 4)
  }
  OUT_WMMA(a0, b00, b01, b02, b03)
  OUT_WMMA(a1, b10, b11, b12, b13)

  // ---- pass 2: query half of W_out (cols H..2H-1) ----
  OUT_LOAD(a2p, a0, b00, b01, b02, b03, 0, H_)
  OUT_LOAD(a2p, a1, b10, b11, b12, b13, 4, H_ + 4)
  for (int k = 8; k < H_; k += 8) {
    OUT_WMMA(a0, b00, b01, b02, b03)
    OUT_LOAD(a2p, a0, b00, b01, b02, b03, k, H_ + k)
    OUT_WMMA(a1, b10, b11, b12, b13)
    OUT_LOAD(a2p, a1, b10, b11, b12, b13, k + 4, H_ + k + 4)
  }
  OUT_WMMA(a0, b00, b01, b02, b03)
  OUT_WMMA(a1, b10, b11, b12, b13)
#undef OUT_LOAD
#undef OUT_WMMA

  v8f* accs[4] = {&acc0, &acc1, &acc2, &acc3};
#pragma unroll
  for (int j = 0; j < 8; ++j) {
    const int row = m0 + j + 8 * hi;
#pragma unroll
    for (int i = 0; i < 4; ++i) {
      const int col = n0 + i * 16 + lo;
      out[(size_t)row * H_ + col] = fast_tanh((*accs[i])[j] + bias[col]);
    }
  }
}

// ---------------------------------------------------------------------------
extern "C" void kernel_launch(void* const* d_in, const int* in_sizes, int n_in,
                              void* d_out, int out_size, void* d_ws, size_t ws_size,
                              hipStream_t stream) {
  (void)in_sizes; (void)n_in; (void)out_size; (void)ws_size;
  const float* query = (const float*)d_in[0];  // (B,T,H)
  const float* enc   = (const float*)d_in[1];  // (B,S,H)
  const int*   lens  = (const int*)d_in[2];    // (B,)
  const float* W_s   = (const float*)d_in[3];  // (H,H)
  const float* W_h   = (const float*)d_in[4];  // (H,H)
  const float* v     = (const float*)d_in[5];  // (H,)
  const float* W_out = (const float*)d_in[6];  // (H,2H)
  const float* b_out = (const float*)d_in[7];  // (H,)
  float* out = (float*)d_out;

  float* ws   = (float*)d_ws;
  float* qs   = ws;                          // B*T*H  = 524288 f
  float* hsbf = qs   + (size_t)B_ * T_ * H_; // B*S*H  = 1048576 f
  float* attn = hsbf + (size_t)B_ * S_ * H_; // B*T*S  = 262144 f
  float* ctx  = attn + (size_t)B_ * T_ * S_; // B*T*H  = 524288 f

  // 1) qs = query @ W_s^T
  linear_wmma<<<dim3(B_ * T_ / 16, H_ / 256), 128, 0, stream>>>(
      query, W_s, qs, B_ * T_, H_, H_);
  // 2) hs = enc @ W_h^T
  linear_wmma<<<dim3(B_ * S_ / 16, H_ / 256), 128, 0, stream>>>(
      enc, W_h, hsbf, B_ * S_, H_, H_);
  // 3) additive scores + masked softmax
  scores_softmax<<<B_ * (T_ / TT), 256, 0, stream>>>(qs, hsbf, v, lens, attn);
  // 4) context = attn @ enc (batched)
  ctx_wmma<<<dim3(T_ / 16, H_ / 256, B_), 128, 0, stream>>>(attn, enc, ctx);
  // 5) out = tanh([ctx, q] @ W_out^T + b_out)
  out_wmma<<<dim3(B_ * T_ / 16, H_ / 256), 128, 0, stream>>>(
      ctx, query, W_out, b_out, out);
}